// L_spa_89550068121954
// MI455X (gfx1250) — compile-verified
//
#include <hip/hip_runtime.h>
#include <stdint.h>

// E[i,j] = sum over 4 dirs of (d[i,j]-d[neighbor])^2,  d = mean_pool(org)-mean_pool(enh)
// org/enh: (32,3,512,512) f32 -> E: (32,1,128,128) f32.
// Bandwidth-bound: 201 MB in / 2 MB out. CDNA5 path: TDM (tensor_load_to_lds) DMA of
// 8x512 f32 tiles into double-buffered LDS, pipelined with TENSORcnt.

#define TPB     256          // 8 wave32 waves
#define POOL_R  16           // pooled output rows per block
#define HALO_R  18
#define NPAIR   9            // pooled-row pairs incl. halo
#define NSTAGE  (NPAIR * 6)  // pairs x 3 channels x {org,enh}

// Dynamic-LDS layout (float indices; dynamic LDS starts at byte offset 0):
//   buf[2][8][512] : double-buffered input tile         (8192 floats, 32 KB)
//   D[18][128]     : pooled diff with 1-row halo        (2304 floats,  9 KB)
#define BUF(bi,row,col)  ((((bi) * 8) + (row)) * 512 + (col))
#define DOFF             8192
#define SMEM_BYTES       ((8192 + 2304) * 4)

#if defined(__HIP_DEVICE_COMPILE__) && \
    __has_builtin(__builtin_amdgcn_tensor_load_to_lds) && \
    __has_builtin(__builtin_amdgcn_s_wait_tensorcnt)
#define USE_TDM 1
#else
#define USE_TDM 0
#endif

#if USE_TDM
typedef unsigned int v4u __attribute__((ext_vector_type(4)));
typedef int          v8i __attribute__((ext_vector_type(8)));
typedef int          v4i __attribute__((ext_vector_type(4)));

// One TDM descriptor: 2D tile, rows of 512 f32 (2048 B), n_rows tall.
// rem_rows = rows remaining in tensor from tile start (rows beyond -> read as zero,
// which implements the reference's zero padding at the bottom edge).
__device__ __forceinline__ void tdm_load_tile(unsigned long long gaddr,
                                              unsigned lds_off,
                                              unsigned rem_rows,
                                              unsigned n_rows)
{
    v4u g0;
    g0.x = 1u;                                                  // count=1, user mode
    g0.y = lds_off;                                             // lds_addr (bytes)
    g0.z = (unsigned)(gaddr & 0xFFFFFFFFull);                   // global_addr[31:0]
    g0.w = (unsigned)((gaddr >> 32) & 0x01FFFFFFull)            // global_addr[56:32]
         | (2u << 30);                                          // type=2 ("image")
    v8i g1;
    g1[0] = (int)(2u << 16);          // workgroup_mask=0 (not in cluster), data_size=4B
    g1[1] = (int)(512u << 16);        // atomic_barrier_addr=0 | tensor_dim0[15:0]=512
    g1[2] = (int)(rem_rows << 16);    // tensor_dim0[31:16]=0  | tensor_dim1[15:0]
    g1[3] = (int)(512u << 16);        // tensor_dim1[31:16]=0  | tile_dim0=512
    g1[4] = (int)n_rows;              // tile_dim1 | tile_dim2=0
    g1[5] = 512;                      // tensor_dim0_stride[31:0] = 512 elements
    g1[6] = 0;                        // stride hi | tensor_dim1_stride lo (unused, 2D)
    g1[7] = 0;

#if __has_builtin(__builtin_amdgcn_tensor_load_to_lds_d2)
    // 2D shorthand: descriptor groups 0 and 1 only (VADDR2/3 = NULL per ISA).
    __builtin_amdgcn_tensor_load_to_lds_d2(g0, g1, 0);
#else
    // amdgpu-toolchain (clang-23) 6-arg form:
    // (uint32x4 g0, int32x8 g1, int32x4, int32x4, int32x8, i32 cpol).
    // Groups 2/3 (and the trailing group) are unused for a 2D tensor -> zeros.
    const v4i zero4 = {0, 0, 0, 0};
    const v8i zero8 = {0, 0, 0, 0, 0, 0, 0, 0};
    __builtin_amdgcn_tensor_load_to_lds(g0, g1, zero4, zero4, zero8, 0);
#endif
}

// Stage k: pair p = k/6 covers pooled rows {r0-1+2p, r0+2p}; r = k%6 picks
// (channel, array). DMA the 8 input rows of that (pair, channel, array).
__device__ __forceinline__ void issue_stage(const float* ob, const float* eb,
                                            int r0, int k)
{
    const int p     = k / 6;
    const int r     = k % 6;
    const int ch    = r >> 1;
    const int arr   = r & 1;
    const int bi    = k & 1;
    const int start = (r0 - 1 + 2 * p) * 4;      // first input row, may be -4
    const int lo    = start < 0 ? 0 : start;     // clamp top edge
    const int skip  = lo - start;                // 0 or 4 rows skipped (halo row -1)
    const unsigned nrows = (unsigned)(8 - skip);
    const unsigned rem   = (unsigned)(512 - lo); // rows left in plane -> OOB reads 0
    const float* src = arr ? eb : ob;
    const unsigned long long ga = (unsigned long long)(uintptr_t)src
        + ((unsigned long long)ch * (512ull * 512ull)
         + (unsigned long long)lo * 512ull) * 4ull;
    const unsigned lds = (unsigned)(BUF(bi, skip, 0) * 4);
    tdm_load_tile(ga, lds, rem, nrows);
}
#endif // USE_TDM

__global__ __launch_bounds__(TPB) void spa_tdm_kernel(
    const float* __restrict__ org,
    const float* __restrict__ enh,
    float* __restrict__ out)
{
    extern __shared__ float smem[];

    const int b    = blockIdx.x >> 3;     // batch 0..31
    const int t    = blockIdx.x & 7;      // row tile 0..7
    const int r0   = t * POOL_R;
    const int half = threadIdx.x >> 7;    // which pooled row of the current pair
    const int pc   = threadIdx.x & 127;   // pooled column

    const size_t chS = (size_t)512 * 512;
    const float* ob = org + (size_t)b * 3 * chS;
    const float* eb = enh + (size_t)b * 3 * chS;

#if USE_TDM
    // ---- Phase 1 (TDM pipeline): pooled diff into D ----
    const bool w0 = (threadIdx.x < 32);   // wave 0 drives the DMA (wave-uniform branch)
    float acc = 0.0f;
    if (w0) issue_stage(ob, eb, r0, 0);
    for (int k = 0; k < NSTAGE; ++k) {
        if (w0) {
            if (k + 1 < NSTAGE) {
                issue_stage(ob, eb, r0, k + 1);          // prefetch next tile
                __builtin_amdgcn_s_wait_tensorcnt(1);    // stage k landed (in-order)
            } else {
                __builtin_amdgcn_s_wait_tensorcnt(0);
            }
        }
        __syncthreads();                                 // publish LDS tile to all waves
        const int bi  = k & 1;
        const int r   = k % 6;
        const int arr = r & 1;
        float s = 0.0f;
        #pragma unroll
        for (int dr = 0; dr < 4; ++dr) {                 // 4 ds_load_b128 per thread
            const float4 v = *(const float4*)&smem[BUF(bi, half * 4 + dr, pc * 4)];
            s += (v.x + v.y) + (v.z + v.w);
        }
        acc += arr ? -s : s;                             // org minus enhance
        if (r == 5) {                                    // pair complete (3 ch x 2 arr)
            const int p  = k / 6;
            const int pr = 2 * p + half;                 // LDS D row 0..17
            const int gr = r0 - 1 + pr;                  // global pooled row
            smem[DOFF + pr * 128 + pc] =
                (gr >= 0 && gr < 128) ? acc * (1.0f / 48.0f) : 0.0f;
            acc = 0.0f;
        }
        __syncthreads();                                 // buffer reusable by stage k+2
    }
#else
    // ---- Fallback: direct global_load_b128 pooling (known-good path) ----
    #pragma unroll
    for (int it = 0; it < HALO_R / 2; ++it) {
        const int pr = it * 2 + half;
        const int gr = r0 - 1 + pr;
        float d = 0.0f;
        if (gr >= 0 && gr < 128) {
            const size_t base = (size_t)(gr * 4) * 512 + (size_t)pc * 4;
            float s = 0.0f;
            #pragma unroll
            for (int ch = 0; ch < 3; ++ch) {
                #pragma unroll
                for (int dr = 0; dr < 4; ++dr) {
                    const size_t off = base + (size_t)ch * chS + (size_t)dr * 512;
                    const float4 vo = *(const float4*)(ob + off);
                    const float4 ve = *(const float4*)(eb + off);
                    s += ((vo.x + vo.y) + (vo.z + vo.w))
                       - ((ve.x + ve.y) + (ve.z + ve.w));
                }
            }
            d = s * (1.0f / 48.0f);
        }
        smem[DOFF + pr * 128 + pc] = d;
    }
    __syncthreads();
#endif

    // ---- Phase 2: E from D ----
    float* ot = out + ((size_t)b * 128 + (size_t)r0) * 128;
    #pragma unroll
    for (int it = 0; it < POOL_R / 2; ++it) {
        const int row = it * 2 + half;                   // 0..15 within tile
        const float dc = smem[DOFF + (row + 1) * 128 + pc];
        const float nl = (pc == 0)   ? 0.0f : smem[DOFF + (row + 1) * 128 + pc - 1];
        const float nr = (pc == 127) ? 0.0f : smem[DOFF + (row + 1) * 128 + pc + 1];
        const float nu = smem[DOFF + row * 128 + pc];    // halo rows are 0 if OOB
        const float nd = smem[DOFF + (row + 2) * 128 + pc];
        const float a0 = dc - nl;
        const float a1 = dc - nr;
        const float a2 = dc - nu;
        const float a3 = dc - nd;
        ot[(size_t)row * 128 + pc] = a0 * a0 + a1 * a1 + a2 * a2 + a3 * a3;
    }
}

extern "C" void kernel_launch(void* const* d_in, const int* in_sizes, int n_in,
                              void* d_out, int out_size, void* d_ws, size_t ws_size,
                              hipStream_t stream) {
    const float* org = (const float*)d_in[0];   // (32,3,512,512) f32
    const float* enh = (const float*)d_in[1];   // (32,3,512,512) f32
    float* out = (float*)d_out;                 // (32,1,128,128) f32

    (void)in_sizes; (void)n_in; (void)out_size; (void)d_ws; (void)ws_size;

    // 32 batches x 8 row-tiles; 42 KB dynamic LDS (buf double-buffer + D).
    spa_tdm_kernel<<<dim3(32 * 8), dim3(TPB), SMEM_BYTES, stream>>>(org, enh, out);
}